// TopKPoolingDiscriminator_63522566308410
// MI455X (gfx1250) — compile-verified
//
#include <hip/hip_runtime.h>
#include <hip/hip_bf16.h>

typedef __attribute__((ext_vector_type(2))) float v2f;
typedef __attribute__((ext_vector_type(8))) float v8f;
typedef unsigned long long u64;

// ---------------------------------------------------------------------------
// 0. zero scratch accumulators (deg[100000] | accOut[200000] | yacc[1])
// ---------------------------------------------------------------------------
__global__ void zero_kernel(float* p, int n) {
    int i = blockIdx.x * blockDim.x + threadIdx.x;
    if (i < n) p[i] = 0.0f;
}

// ---------------------------------------------------------------------------
// 1. h = x @ conv_w  via V_WMMA_F32_16X16X4_F32.
//    One wave -> one 16-row tile; K=128 in steps of 4 (32 wmma ops).
//    A (16x4 f32, 2 VGPR):  lane l: M=l&15, VGPR j holds K = ((l>=16)?2:0)+j
//    B (4x16 f32, 2 VGPR):  lane l: N=l&15, VGPR j holds K = ((l>=16)?2:0)+j
//    C/D (16x16 f32, 8 VGPR): VGPR v: M = v + ((l>=16)?8:0), N = l&15
// ---------------------------------------------------------------------------
__global__ __launch_bounds__(128) void gemm_wmma(const float* __restrict__ x,
                                                 const float* __restrict__ w,
                                                 float* __restrict__ h, int n) {
    const int lane = threadIdx.x & 31;
    const int wave = threadIdx.x >> 5;
    const int tile = blockIdx.x * 4 + wave;
    const int row0 = tile * 16;
    if (row0 >= n) return;                      // uniform per-wave exit

    const int m    = lane & 15;
    const int kb   = (lane >> 4) << 1;          // 0 or 2
    const int colN = lane & 15;
    const int cw   = (colN < 2) ? colN : 0;     // clamp (branchless)
    const float bm = (colN < 2) ? 1.0f : 0.0f;  // zero pad cols >= DIM

    int xr = row0 + m; if (xr >= n) xr = n - 1; // clamp (never stored anyway)
    const float* xrow = x + (size_t)xr * 128 + kb;

    v8f c = {};
    for (int k0 = 0; k0 < 128; k0 += 4) {
        v2f a, b;
        a.x = xrow[k0];
        a.y = xrow[k0 + 1];
        b.x = w[(k0 + kb) * 2 + cw] * bm;
        b.y = w[(k0 + kb + 1) * 2 + cw] * bm;
        c = __builtin_amdgcn_wmma_f32_16x16x4_f32(
                /*neg_a=*/false, a, /*neg_b=*/false, b,
                /*c_mod=*/(short)0, c, /*reuse_a=*/false, /*reuse_b=*/false);
    }

    if (colN < 2) {
        const int mh = (lane >> 4) << 3;        // +8 for upper lane half
        #pragma unroll
        for (int v = 0; v < 8; ++v) {
            int r = row0 + v + mh;
            if (r < n) h[(size_t)r * 2 + colN] = c[v];
        }
    }
}

// ---------------------------------------------------------------------------
// 2. deg = segment_sum(edge_attr, col) + 1 per node (self loops)
// ---------------------------------------------------------------------------
__global__ void deg_kernel(const long long* __restrict__ col,
                           const float* __restrict__ ea,
                           float* __restrict__ deg, int E, int n) {
    int e = blockIdx.x * blockDim.x + threadIdx.x;
    if (e < E) {
        atomicAdd(&deg[(int)col[e]], ea[e]);
    } else if (e < E + n) {
        atomicAdd(&deg[e - E], 1.0f);
    }
}

__global__ void dinv_kernel(const float* __restrict__ deg,
                            float* __restrict__ dinv, int n) {
    int i = blockIdx.x * blockDim.x + threadIdx.x;
    if (i < n) {
        float d = deg[i];
        dinv[i] = (d > 0.0f) ? rsqrtf(d) : 0.0f;
    }
}

// ---------------------------------------------------------------------------
// 3. acc[col] += dinv[row]*w*dinv[col] * h[row]   (+ self-loop messages)
// ---------------------------------------------------------------------------
__global__ void scatter_kernel(const long long* __restrict__ row,
                               const long long* __restrict__ col,
                               const float* __restrict__ ea,
                               const float* __restrict__ dinv,
                               const float* __restrict__ h,
                               float* __restrict__ acc, int E, int n) {
    int e = blockIdx.x * blockDim.x + threadIdx.x;
    if (e < E) {
        int r = (int)row[e], c = (int)col[e];
        float nrm = dinv[r] * ea[e] * dinv[c];
        atomicAdd(&acc[2 * c],     nrm * h[2 * r]);
        atomicAdd(&acc[2 * c + 1], nrm * h[2 * r + 1]);
    } else if (e < E + n) {
        int i = e - E;
        float nrm = dinv[i] * dinv[i];
        atomicAdd(&acc[2 * i],     nrm * h[2 * i]);
        atomicAdd(&acc[2 * i + 1], nrm * h[2 * i + 1]);
    }
}

// ---------------------------------------------------------------------------
// 4. relu + bias, score = tanh(out.pool_w/||pool_w||), build unique sort key
//    key = (~ascending_map(score) << shift) | i  -> ascending key order
//    == descending score with stable tie-break by index.
// ---------------------------------------------------------------------------
__global__ void score_kernel(float* __restrict__ acc,
                             const float* __restrict__ conv_b,
                             const float* __restrict__ pool_w,
                             float* __restrict__ score,
                             u64* __restrict__ keys,
                             int n, int npad, int shift) {
    int i = blockIdx.x * blockDim.x + threadIdx.x;
    if (i < n) {
        float o0 = fmaxf(acc[2 * i] + conv_b[0], 0.0f);
        float o1 = fmaxf(acc[2 * i + 1] + conv_b[1], 0.0f);
        acc[2 * i] = o0;
        acc[2 * i + 1] = o1;
        float pw0 = pool_w[0], pw1 = pool_w[1];
        float inv = rsqrtf(pw0 * pw0 + pw1 * pw1);
        float s = tanhf((o0 * pw0 + o1 * pw1) * inv);
        score[i] = s;
        unsigned u = __float_as_uint(s);
        u = (u & 0x80000000u) ? ~u : (u | 0x80000000u);   // ascending float map
        unsigned d = ~u;                                  // descending
        keys[i] = ((u64)d << shift) | (unsigned)i;
    } else if (i < npad) {
        keys[i] = ~0ULL;                                  // pad to the end
    }
}

// ---------------------------------------------------------------------------
// 5. bitonic sort step (ascending), launched per (k, j)
// ---------------------------------------------------------------------------
__global__ void bitonic_step(u64* __restrict__ keys, int j, int k) {
    int i = blockIdx.x * blockDim.x + threadIdx.x;
    int ixj = i ^ j;
    if (ixj > i) {
        u64 a = keys[i], b = keys[ixj];
        bool up = ((i & k) == 0);
        if ((a > b) == up) { keys[i] = b; keys[ixj] = a; }
    }
}

// ---------------------------------------------------------------------------
// 6. y_acc = sum_p score[i(p)] * (out[i,0]*fc_w[2p] + out[i,1]*fc_w[2p+1])
// ---------------------------------------------------------------------------
__global__ void dot_kernel(const u64* __restrict__ keys,
                           const float* __restrict__ acc,
                           const float* __restrict__ score,
                           const float* __restrict__ fc_w,
                           float* __restrict__ yacc, int n, u64 idx_mask) {
    __shared__ float red[256];
    int p = blockIdx.x * blockDim.x + threadIdx.x;
    float v = 0.0f;
    if (p < n) {
        int i = (int)(keys[p] & idx_mask);
        float s = score[i];
        v = s * (acc[2 * i] * fc_w[2 * p] + acc[2 * i + 1] * fc_w[2 * p + 1]);
    }
    red[threadIdx.x] = v;
    __syncthreads();
    for (int s = 128; s > 0; s >>= 1) {
        if ((int)threadIdx.x < s) red[threadIdx.x] += red[threadIdx.x + s];
        __syncthreads();
    }
    if (threadIdx.x == 0) atomicAdd(yacc, red[0]);
}

__global__ void final_kernel(const float* __restrict__ yacc,
                             const float* __restrict__ fc_b,
                             float* __restrict__ out) {
    if (threadIdx.x == 0) {
        float z = yacc[0] + fc_b[0];
        out[0] = 1.0f / (1.0f + expf(-z));
    }
}

// ---------------------------------------------------------------------------
// launch
// ---------------------------------------------------------------------------
extern "C" void kernel_launch(void* const* d_in, const int* in_sizes, int n_in,
                              void* d_out, int out_size, void* d_ws, size_t ws_size,
                              hipStream_t stream) {
    const float*     x      = (const float*)d_in[0];
    const long long* eidx   = (const long long*)d_in[1];  // int64 [2, E]
    const float*     eattr  = (const float*)d_in[2];
    const float*     conv_w = (const float*)d_in[3];
    const float*     conv_b = (const float*)d_in[4];
    const float*     pool_w = (const float*)d_in[5];
    const float*     fc_w   = (const float*)d_in[6];
    const float*     fc_b   = (const float*)d_in[7];
    float*           out    = (float*)d_out;

    const int n = in_sizes[0] / 128;     // 100000
    const int E = in_sizes[2];           // 3200000
    const long long* row = eidx;
    const long long* col = eidx + E;

    // scratch layout (floats):
    float* ws     = (float*)d_ws;
    float* deg    = ws;                  // n
    float* acc    = ws + n;              // 2n
    float* yacc   = ws + 3 * n;          // 1
    float* h      = ws + 3 * n + 4;      // 2n
    float* dinv   = ws + 5 * n + 4;      // n
    float* score  = ws + 6 * n + 4;      // n
    u64*   keys   = (u64*)(ws + 7 * n + 8);   // npad (8-byte aligned)

    int shift = 0; while ((1 << shift) < n) ++shift;        // 17
    int npad  = 1;  while (npad < n) npad <<= 1;            // 131072
    const u64 idx_mask = ((u64)1 << shift) - 1;

    const int B = 256;

    // 0. zero accumulators: deg (n) + acc (2n) + yacc (1) are contiguous
    {
        int tot = 3 * n + 1;
        zero_kernel<<<(tot + B - 1) / B, B, 0, stream>>>(ws, tot);
    }
    // 1. h = x @ conv_w  (WMMA f32)
    {
        int tiles = (n + 15) / 16;
        int blocks = (tiles + 3) / 4;    // 4 waves/block, 1 tile/wave
        gemm_wmma<<<blocks, 128, 0, stream>>>(x, conv_w, h, n);
    }
    // 2. degrees + dinv
    {
        int tot = E + n;
        deg_kernel<<<(tot + B - 1) / B, B, 0, stream>>>(col, eattr, deg, E, n);
        dinv_kernel<<<(n + B - 1) / B, B, 0, stream>>>(deg, dinv, n);
    }
    // 3. message scatter
    {
        int tot = E + n;
        scatter_kernel<<<(tot + B - 1) / B, B, 0, stream>>>(row, col, eattr,
                                                            dinv, h, acc, E, n);
    }
    // 4. relu/bias/score/keys
    score_kernel<<<(npad + B - 1) / B, B, 0, stream>>>(acc, conv_b, pool_w,
                                                       score, keys, n, npad, shift);
    // 5. bitonic sort (ascending unique keys == stable descending-score argsort)
    {
        int blocks = (npad + B - 1) / B;
        for (int k = 2; k <= npad; k <<= 1)
            for (int j = k >> 1; j > 0; j >>= 1)
                bitonic_step<<<blocks, B, 0, stream>>>(keys, j, k);
    }
    // 6. permuted dot with fc_w, then sigmoid
    dot_kernel<<<(n + B - 1) / B, B, 0, stream>>>(keys, acc, score, fc_w,
                                                  yacc, n, idx_mask);
    final_kernel<<<1, 32, 0, stream>>>(yacc, fc_b, out);
}